// RecallCELoss_35734127902813
// MI455X (gfx1250) — compile-verified
//
#include <hip/hip_runtime.h>
#include <hip/hip_bf16.h>
#include <cstdint>

#define NUM_CLASSES 7
#define BLOCK 256
#define WAVE 32
#define TILE_ROWS 1024
#define TILE_DWORDS (TILE_ROWS * NUM_CLASSES)   // 7168 dwords = 28 KB LDS

// ---------------- gfx1250 async global->LDS helpers ----------------

typedef int vint4 __attribute__((vector_size(4 * sizeof(int))));
typedef __attribute__((address_space(3))) vint4 lds_vint4;
typedef __attribute__((address_space(3))) int   lds_int;

__device__ __forceinline__ void async_load_b128(const float* gaddr, void* lds) {
#if __has_builtin(__builtin_amdgcn_global_load_async_to_lds_b128)
    __builtin_amdgcn_global_load_async_to_lds_b128(
        (vint4*)gaddr, (lds_vint4*)lds, /*offset=*/0, /*cpol=*/0);
#else
    unsigned lds_off =
        (unsigned)(unsigned long long)(__attribute__((address_space(3))) char*)lds;
    asm volatile("global_load_async_to_lds_b128 %0, %1, off"
                 :: "v"(lds_off), "v"(gaddr) : "memory");
#endif
}

__device__ __forceinline__ void async_load_b32(const float* gaddr, void* lds) {
#if __has_builtin(__builtin_amdgcn_global_load_async_to_lds_b32)
    __builtin_amdgcn_global_load_async_to_lds_b32(
        (int*)gaddr, (lds_int*)lds, /*offset=*/0, /*cpol=*/0);
#else
    unsigned lds_off =
        (unsigned)(unsigned long long)(__attribute__((address_space(3))) char*)lds;
    asm volatile("global_load_async_to_lds_b32 %0, %1, off"
                 :: "v"(lds_off), "v"(gaddr) : "memory");
#endif
}

__device__ __forceinline__ void wait_async_zero() {
#if __has_builtin(__builtin_amdgcn_s_wait_asynccnt)
    __builtin_amdgcn_s_wait_asynccnt(0);
#else
    asm volatile("s_wait_asynccnt 0" ::: "memory");
#endif
}

__device__ __forceinline__ unsigned ballot32(bool p) {
#if __has_builtin(__builtin_amdgcn_ballot_w32)
    return __builtin_amdgcn_ballot_w32(p);
#else
    return (unsigned)__ballot(p);
#endif
}

// ---------------- kernels ----------------

__global__ void recall_init_kernel(unsigned* gcnt) {
    if (threadIdx.x < 2 * NUM_CLASSES) gcnt[threadIdx.x] = 0u;
}

// gcnt layout: [0..6] = tp[c], [7..13] = nc[c]
__global__ __launch_bounds__(BLOCK)
void recall_hist_kernel(const float* __restrict__ inputs,
                        const int*   __restrict__ targets,
                        unsigned*    __restrict__ gcnt,
                        int n) {
    __shared__ float    tile[TILE_DWORDS];
    __shared__ unsigned hist[2 * NUM_CLASSES];

    const int tid  = threadIdx.x;
    const int lane = tid & (WAVE - 1);
    if (tid < 2 * NUM_CLASSES) hist[tid] = 0u;

    const long long rowBase = (long long)blockIdx.x * TILE_ROWS;
    long long remRows = (long long)n - rowBase;
    const int rows = (remRows < TILE_ROWS) ? (int)remRows : TILE_ROWS;
    if (rows <= 0) return;

    const int dwords = rows * NUM_CLASSES;
    const float* gsrc = inputs + rowBase * NUM_CLASSES;

    // ---- Stream the tile into LDS with per-lane 128-bit async transfers ----
    const int nChunks = dwords >> 2;
    for (int c = tid; c < nChunks; c += BLOCK) {
        async_load_b128(gsrc + 4 * c, (char*)tile + 16 * c);
    }
    const int rem = dwords & 3;                  // ragged tail (tail tile only)
    if (tid < rem) {
        const int d = (nChunks << 2) + tid;
        async_load_b32(gsrc + d, (char*)tile + 4 * d);
    }

    wait_async_zero();      // this wave's async ops landed in LDS
    __syncthreads();        // all waves' tiles visible

    // ---- Argmax + wave32 ballot histogram (no per-row atomics) ----
    unsigned tp_acc[NUM_CLASSES];
    unsigned nc_acc[NUM_CLASSES];
#pragma unroll
    for (int c = 0; c < NUM_CLASSES; ++c) { tp_acc[c] = 0u; nc_acc[c] = 0u; }

    const int iters = (rows + BLOCK - 1) / BLOCK;   // uniform trip count
    for (int j = 0; j < iters; ++j) {
        const int r = tid + j * BLOCK;              // r < TILE_ROWS always
        const bool valid = (r < rows);

        // LDS row read: lane stride = 7 dwords, gcd(7,64)=1 -> conflict-free.
        const float* row = &tile[r * NUM_CLASSES];
        float m = row[0];
        int arg = 0;
#pragma unroll
        for (int c = 1; c < NUM_CLASSES; ++c) {
            float v = row[c];
            if (v > m) { m = v; arg = c; }          // strict '>': first-max, like argmax
        }
        const int t = valid ? targets[rowBase + r] : -1;   // coalesced
        const bool correct = valid && (arg == t);

#pragma unroll
        for (int c = 0; c < NUM_CLASSES; ++c) {
            nc_acc[c] += __popc(ballot32(t == c));
            tp_acc[c] += __popc(ballot32(correct && (t == c)));
        }
    }

    // One set of LDS atomics per wave (lane 0 holds full wave totals).
    if (lane == 0) {
#pragma unroll
        for (int c = 0; c < NUM_CLASSES; ++c) {
            if (tp_acc[c]) atomicAdd(&hist[c], tp_acc[c]);
            if (nc_acc[c]) atomicAdd(&hist[NUM_CLASSES + c], nc_acc[c]);
        }
    }

    __syncthreads();
    if (tid < 2 * NUM_CLASSES) {
        unsigned v = hist[tid];
        if (v) atomicAdd(&gcnt[tid], v);
    }
}

__global__ void recall_finalize_kernel(const unsigned* __restrict__ gcnt,
                                       float* __restrict__ out) {
    if (threadIdx.x == 0) {
        float s = 0.0f;
#pragma unroll
        for (int c = 0; c < NUM_CLASSES; ++c) {
            float tp = (float)gcnt[c];
            float nc = (float)gcnt[NUM_CLASSES + c];
            float recall = tp / (nc + 1e-8f);
            const float pc = 1.0f;
            s += (1.0f - recall) * nc * __logf(pc);   // log(1) == 0, faithful to ref
        }
        out[0] = -(s / (float)NUM_CLASSES);
    }
}

// ---------------- launch ----------------

extern "C" void kernel_launch(void* const* d_in, const int* in_sizes, int n_in,
                              void* d_out, int out_size, void* d_ws, size_t ws_size,
                              hipStream_t stream) {
    const float* inputs  = (const float*)d_in[0];
    const int*   targets = (const int*)d_in[1];
    const int    n       = in_sizes[1];          // N rows (targets count)
    unsigned*    gcnt    = (unsigned*)d_ws;      // 14 counters

    recall_init_kernel<<<1, 32, 0, stream>>>(gcnt);

    const int tiles = (n + TILE_ROWS - 1) / TILE_ROWS;
    recall_hist_kernel<<<tiles, BLOCK, 0, stream>>>(inputs, targets, gcnt, n);

    recall_finalize_kernel<<<1, 32, 0, stream>>>(gcnt, (float*)d_out);
}